// SplineCNN_42099269435912
// MI455X (gfx1250) — compile-verified
//
#include <hip/hip_runtime.h>
#include <hip/hip_bf16.h>

typedef float v2f __attribute__((ext_vector_type(2)));
typedef float v8f __attribute__((ext_vector_type(8)));

#define KSZ   5            // kernel_size per pseudo-dim
#define KK    25           // 5*5 kernel matrices
#define C_OUT 16           // out channels per layer (H == Cout == 16)

// ---------------------------------------------------------------------------
// Precompute y[n, k, :] = x[n, :] @ W[k, :, :]  for all 25 kernels.
// Exact f32 math via V_WMMA_F32_16X16X4_F32 (wave32, 16x16 tile, K-step 4).
// All 25 weight matrices staged in LDS (25*CIN*16*4 B: 100 KB for CIN=64,
// fits CDNA5's 320 KB WGP LDS). One wave owns one 16-node tile at a time.
// Full tiles take an unguarded store path (wave-uniform branch) so the hot
// k-loop has no per-element exec-mask manipulation.
// ---------------------------------------------------------------------------
template<int CIN>
__global__ __launch_bounds__(256)
void precompute_y_kernel(const float* __restrict__ x,   // [N, CIN]
                         const float* __restrict__ W,   // [25, CIN, 16]
                         float* __restrict__ y,         // [N, 25, 16]
                         int N)
{
    __shared__ float sW[KK * CIN * C_OUT];
    for (int i = threadIdx.x; i < KK * CIN * C_OUT; i += blockDim.x) sW[i] = W[i];
    __syncthreads();

    const int lane = threadIdx.x & 31;
    const int wave = threadIdx.x >> 5;
    const int wavesPerBlk = blockDim.x >> 5;
    const int m  = lane & 15;   // A: row M within tile; B: output channel N
    const int kp = lane >> 4;   // which K-pair within the 4-wide K step

    const int tiles = (N + 15) >> 4;
    for (int tile = blockIdx.x * wavesPerBlk + wave; tile < tiles;
         tile += gridDim.x * wavesPerBlk) {
        const int node0 = tile << 4;
        const bool fullTile = (node0 + 16 <= N);   // wave-uniform

        // A-tile (16 x CIN) in WMMA A layout: lane holds (M=m, K=4s+2*kp+{0,1})
        v2f a[CIN / 4];
        int row = node0 + m; if (row >= N) row = N - 1;   // clamp; stores guarded
        const float* xr = x + (size_t)row * CIN;
        #pragma unroll
        for (int s = 0; s < CIN / 4; ++s) {
            a[s].x = xr[4 * s + 2 * kp + 0];
            a[s].y = xr[4 * s + 2 * kp + 1];
        }

        // Per-lane store base: this lane's first output element is
        // y[(node0 + 8*kp) * 400 + m]; row r adds 400 floats, kernel k adds 16.
        float* ybase = y + (size_t)(node0 + 8 * kp) * (KK * C_OUT) + m;

        for (int k = 0; k < KK; ++k) {
            const float* wk = sW + k * CIN * C_OUT;
            v8f c = {};
            #pragma unroll
            for (int s = 0; s < CIN / 4; ++s) {
                v2f b;  // B layout mirrors A: (K=4s+2*kp+{0,1}, N=m)
                b.x = wk[(4 * s + 2 * kp + 0) * C_OUT + m];
                b.y = wk[(4 * s + 2 * kp + 1) * C_OUT + m];
                c = __builtin_amdgcn_wmma_f32_16x16x4_f32(
                        false, a[s], false, b, (short)0, c, false, false);
            }
            // C/D layout: VGPR r -> M = r + 8*kp, N = m.
            // Lanes 0-15 write 64 B contiguous (one node's 16 channels of
            // kernel k), lanes 16-31 another 64 B: fully coalesced.
            if (fullTile) {
                #pragma unroll
                for (int r = 0; r < 8; ++r)
                    ybase[(size_t)r * (KK * C_OUT) + k * C_OUT] = c[r];
            } else {
                #pragma unroll
                for (int r = 0; r < 8; ++r) {
                    int node = node0 + r + 8 * kp;
                    if (node < N)
                        y[(size_t)node * (KK * C_OUT) + k * C_OUT + m] = c[r];
                }
            }
        }
    }
}

// ---------------------------------------------------------------------------
// Edge pass: 16 lanes per edge (one lane per output channel).
// 4 B-spline taps, each a coalesced 64 B gather from the L2-resident y table,
// then global_atomic_add_f32 scatter into agg[dst] (+ degree count).
// ---------------------------------------------------------------------------
__global__ __launch_bounds__(256)
void edge_kernel(const float* __restrict__ y,       // [N, 25, 16]
                 const int*   __restrict__ esrc,    // [E]
                 const int*   __restrict__ edst,    // [E]
                 const float* __restrict__ pseudo,  // [E, 2]
                 float* __restrict__ agg,           // [N, 16]
                 float* __restrict__ deg,           // [N]
                 int E)
{
    const long long t = (long long)blockIdx.x * blockDim.x + threadIdx.x;
    const int e = (int)(t >> 4);
    const int c = (int)(t & 15);
    if (e >= E) return;

    const float u0 = pseudo[2 * e + 0];
    const float u1 = pseudo[2 * e + 1];
    const float v0 = u0 * (float)(KSZ - 1);
    const float v1 = u1 * (float)(KSZ - 1);
    const float fl0 = floorf(v0), fl1 = floorf(v1);
    const float f0 = v0 - fl0,    f1 = v1 - fl1;
    const int   i0 = (int)fl0,    i1 = (int)fl1;

    const int src = esrc[e];
    const int dst = edst[e];
    const float* yb = y + (size_t)src * (KK * C_OUT);

    float msg = 0.0f;
    #pragma unroll
    for (int s0 = 0; s0 < 2; ++s0) {
        const float b0 = s0 ? f0 : (1.0f - f0);
        const int   j0 = (i0 + s0) % KSZ;
        #pragma unroll
        for (int s1 = 0; s1 < 2; ++s1) {
            const float b1 = s1 ? f1 : (1.0f - f1);
            const int   j1 = (i1 + s1) % KSZ;
            msg = fmaf(b0 * b1, yb[(j0 + j1 * KSZ) * C_OUT + c], msg);
        }
    }
    atomicAdd(&agg[(size_t)dst * C_OUT + c], msg);
    if (c == 0) atomicAdd(&deg[dst], 1.0f);
}

// ---------------------------------------------------------------------------
// Finalize: out = agg / max(deg,1) + x @ W_root + bias  (+ optional ReLU)
// ---------------------------------------------------------------------------
template<int CIN, bool RELU>
__global__ __launch_bounds__(256)
void finalize_kernel(const float* __restrict__ agg,   // [N, 16]
                     const float* __restrict__ deg,   // [N]
                     const float* __restrict__ x,     // [N, CIN]
                     const float* __restrict__ root,  // [CIN, 16]
                     const float* __restrict__ bias,  // [16]
                     float* __restrict__ out,         // [N, 16]
                     int N)
{
    const long long t = (long long)blockIdx.x * blockDim.x + threadIdx.x;
    const int n = (int)(t >> 4);
    const int c = (int)(t & 15);
    if (n >= N) return;

    float acc = bias[c];
    const float* xr = x + (size_t)n * CIN;
    #pragma unroll 8
    for (int i = 0; i < CIN; ++i)
        acc = fmaf(xr[i], root[i * C_OUT + c], acc);

    float d = deg[n];
    d = d > 1.0f ? d : 1.0f;
    float r = agg[(size_t)n * C_OUT + c] / d + acc;
    if (RELU) r = r > 0.0f ? r : 0.0f;
    out[(size_t)n * C_OUT + c] = r;
}

// ---------------------------------------------------------------------------
// Host orchestration.
// Inputs (setup_inputs order): x, edge_index, pseudo, W1, root1, b1, W2, root2, b2
// ---------------------------------------------------------------------------
extern "C" void kernel_launch(void* const* d_in, const int* in_sizes, int n_in,
                              void* d_out, int out_size, void* d_ws, size_t ws_size,
                              hipStream_t stream) {
    const float* x      = (const float*)d_in[0];
    const int*   eidx   = (const int*)  d_in[1];
    const float* pseudo = (const float*)d_in[2];
    const float* W1     = (const float*)d_in[3];
    const float* root1  = (const float*)d_in[4];
    const float* b1     = (const float*)d_in[5];
    const float* W2     = (const float*)d_in[6];
    const float* root2  = (const float*)d_in[7];
    const float* b2     = (const float*)d_in[8];
    float*       out    = (float*)d_out;

    const int N = in_sizes[0] / 64;   // Cin = 64
    const int E = in_sizes[1] / 2;    // edge_index is [2, E]
    const int* esrc = eidx;
    const int* edst = eidx + E;

    // Workspace carve-up (all f32): y | agg | deg | h
    float* y   = (float*)d_ws;                         // N*25*16  (160 MB, reused)
    float* agg = y + (size_t)N * KK * C_OUT;           // N*16
    float* deg = agg + (size_t)N * C_OUT;              // N
    float* h   = deg + (size_t)N;                      // N*16

    const int tiles   = (N + 15) / 16;
    const int gemmBlk = (tiles + 7) / 8;               // 8 waves / 256-thread block
    const int edgeBlk = (int)(((long long)E * 16 + 255) / 256);
    const int nodeBlk = (int)(((long long)N * 16 + 255) / 256);

    // ---------------- Layer 1: Cin=64 -> 16, ReLU ----------------
    precompute_y_kernel<64><<<gemmBlk, 256, 0, stream>>>(x, W1, y, N);
    hipMemsetAsync(agg, 0, (size_t)N * (C_OUT + 1) * sizeof(float), stream); // agg+deg contiguous
    edge_kernel<<<edgeBlk, 256, 0, stream>>>(y, esrc, edst, pseudo, agg, deg, E);
    finalize_kernel<64, true><<<nodeBlk, 256, 0, stream>>>(agg, deg, x, root1, b1, h, N);

    // ---------------- Layer 2: Cin=16 -> 16 ----------------------
    precompute_y_kernel<16><<<gemmBlk, 256, 0, stream>>>(h, W2, y, N);
    hipMemsetAsync(agg, 0, (size_t)N * (C_OUT + 1) * sizeof(float), stream);
    edge_kernel<<<edgeBlk, 256, 0, stream>>>(y, esrc, edst, pseudo, agg, deg, E);
    finalize_kernel<16, false><<<nodeBlk, 256, 0, stream>>>(agg, deg, h, root2, b2, out, N);
}